// BipartiteGNN_69337952027252
// MI455X (gfx1250) — compile-verified
//
#include <hip/hip_runtime.h>

#define NP 10000           // plant nodes
#define NT 20000           // total nodes
#define NE 640000          // edges
#define HD 128             // feature / hidden dim

typedef float v2f __attribute__((ext_vector_type(2)));
typedef float v8f __attribute__((ext_vector_type(8)));

// ---------------------------------------------------------------------------
// CDNA5 async global->LDS copy (ASYNCcnt-tracked), 16B per lane.
// vdst operand is the 32-bit LDS byte address (low 32 bits of the flat VA).
// ---------------------------------------------------------------------------
__device__ __forceinline__ void async_g2l_b128(unsigned lds_off,
                                               const void* gptr) {
  asm volatile("global_load_async_to_lds_b128 %0, %1, off"
               :: "v"(lds_off), "v"(gptr)
               : "memory");
}
__device__ __forceinline__ void wait_async0() {
  asm volatile("s_wait_asynccnt 0x0" ::: "memory");
}

// ---------------------------------------------------------------------------
// One 16x16 output tile of (A[16x128] @ W[128x128]) accumulated via
// V_WMMA_F32_16X16X4_F32.  A comes from LDS (padded stride), W from global
// (128x128 row-major, resident in L2).
//
// Fragment layouts (CDNA5 ISA 7.12.2, wave32):
//   A 16x4 f32 : vgpr0 = K0 (lanes 0-15) / K2 (lanes 16-31), vgpr1 = K1/K3
//   B 4x16 f32 : vgpr0 = row K0 (lanes 0-15) / row K2 (lanes 16-31), vgpr1 = K1/K3
//   C/D 16x16  : vgpr v = row (v + 8*(lane>=16)), col = lane&15
// ---------------------------------------------------------------------------
__device__ __forceinline__ v8f wmma_tile_f32(const float (*A)[HD + 4],
                                             const float* __restrict__ W,
                                             int n0, int lane, v8f acc) {
  const int m    = lane & 15;        // A row / B-D col within tile
  const int koff = (lane >> 4) << 1; // 0 for lanes 0-15, 2 for lanes 16-31
#pragma unroll
  for (int k0 = 0; k0 < HD; k0 += 4) {
    const int k = k0 + koff;
    v2f a, b;
    a.x = A[m][k];
    a.y = A[m][k + 1];
    b.x = W[(k + 0) * HD + n0 + m];
    b.y = W[(k + 1) * HD + n0 + m];
    acc = __builtin_amdgcn_wmma_f32_16x16x4_f32(false, a, false, b,
                                                (short)0, acc, false, false);
  }
  return acc;
}

// ---------------------------------------------------------------------------
// Fused per-type 2-layer MLP: h0[row] = relu(x@W1 + b1) @ W2 + b2
// One block = one 16-row tile; 8 waves cover the 128 output columns.
// Tiles [0,625) are plants, [625,1250) are pollinators.
// ---------------------------------------------------------------------------
__global__ __launch_bounds__(256) void mlp128_kernel(
    const float* __restrict__ x,
    const float* __restrict__ pW1, const float* __restrict__ pb1,
    const float* __restrict__ pW2, const float* __restrict__ pb2,
    const float* __restrict__ qW1, const float* __restrict__ qb1,
    const float* __restrict__ qW2, const float* __restrict__ qb2,
    float* __restrict__ h0) {
  __shared__ float tA[16][HD + 4];
  __shared__ float tT[16][HD + 4];

  const int  tile  = blockIdx.x;
  const bool plant = tile < (NP / 16);
  const float* W1 = plant ? pW1 : qW1;
  const float* b1 = plant ? pb1 : qb1;
  const float* W2 = plant ? pW2 : qW2;
  const float* b2 = plant ? pb2 : qb2;

  const int row0 = tile * 16;
  const int tid  = threadIdx.x;

  // async-stage the 16x128 input tile into LDS (padded rows; 16B chunks
  // never straddle a row boundary).
  const unsigned ldsA = (unsigned)(uintptr_t)&tA[0][0];
  for (int i = tid * 4; i < 16 * HD; i += 256 * 4) {
    const unsigned loff = ldsA + (unsigned)(((i >> 7) * (HD + 4) + (i & 127)) * 4);
    async_g2l_b128(loff, x + row0 * HD + i);
  }
  wait_async0();
  __syncthreads();

  const int lane = tid & 31;
  const int n0   = (tid >> 5) * 16;
  const int m    = lane & 15;
  const int half = lane >> 4;

  v8f acc = {};
  acc = wmma_tile_f32(tA, W1, n0, lane, acc);
  const float bias1 = b1[n0 + m];
#pragma unroll
  for (int v = 0; v < 8; ++v)
    tT[v + 8 * half][n0 + m] = fmaxf(acc[v] + bias1, 0.0f);
  __syncthreads();

  v8f acc2 = {};
  acc2 = wmma_tile_f32(tT, W2, n0, lane, acc2);
  const float bias2 = b2[n0 + m];
#pragma unroll
  for (int v = 0; v < 8; ++v)
    h0[(row0 + v + 8 * half) * HD + n0 + m] = acc2[v] + bias2;
}

// ---------------------------------------------------------------------------
// hw = in @ W   (M x 128 @ 128 x 128), one 16-row tile per block.
// ---------------------------------------------------------------------------
__global__ __launch_bounds__(256) void gemm128_kernel(
    const float* __restrict__ in, const float* __restrict__ W,
    float* __restrict__ out) {
  __shared__ float tA[16][HD + 4];
  const int row0 = blockIdx.x * 16;
  const int tid  = threadIdx.x;

  const unsigned ldsA = (unsigned)(uintptr_t)&tA[0][0];
  for (int i = tid * 4; i < 16 * HD; i += 256 * 4) {
    const unsigned loff = ldsA + (unsigned)(((i >> 7) * (HD + 4) + (i & 127)) * 4);
    async_g2l_b128(loff, in + row0 * HD + i);
  }
  wait_async0();
  __syncthreads();

  const int lane = tid & 31;
  const int n0   = (tid >> 5) * 16;

  v8f acc = {};
  acc = wmma_tile_f32(tA, W, n0, lane, acc);

  const int m    = lane & 15;
  const int half = lane >> 4;
#pragma unroll
  for (int v = 0; v < 8; ++v)
    out[(row0 + v + 8 * half) * HD + n0 + m] = acc[v];
}

// ---------------------------------------------------------------------------
// Helpers: fill, degree accumulation, dinv, edge scatter, finalize.
// ---------------------------------------------------------------------------
__global__ void fill_kernel(float* __restrict__ p, float v, int n) {
  int t = blockIdx.x * blockDim.x + threadIdx.x;
  if (t < n) p[t] = v;
}

__global__ void deg_kernel(const long long* __restrict__ ei,
                           float* __restrict__ deg) {
  int e = blockIdx.x * blockDim.x + threadIdx.x;
  if (e < NE) atomicAdd(&deg[(int)ei[NE + e]], 1.0f);
}

__global__ void dinv_kernel(const float* __restrict__ deg,
                            float* __restrict__ dinv) {
  int t = blockIdx.x * blockDim.x + threadIdx.x;
  if (t < NT) dinv[t] = __frsqrt_rn(deg[t]);  // deg >= 1 (self loop)
}

// 32 lanes per edge; each lane moves 4 floats.  The edge index is
// wave-uniform -> scalarize it so ei/dinv reads become SMEM loads.
// hw[src] gather uses TH_LOAD_NT_RT: skip the 64KB WGP$ (no reuse there),
// keep normal policy in the 192MB L2 where hw (10MB) is resident.
__global__ __launch_bounds__(256) void scatter_kernel(
    const long long* __restrict__ ei, const float* __restrict__ hw,
    const float* __restrict__ dinv, float* __restrict__ agg) {
  const int t = blockIdx.x * blockDim.x + threadIdx.x;
  const int e = __builtin_amdgcn_readfirstlane(t >> 5);  // wave-uniform edge
  if (e >= NE) return;
  const int c = (t & 31) * 4;
  const int s = (int)ei[e];
  const int d = (int)ei[NE + e];
  const float w = dinv[s] * dinv[d];

  float4 v;
  const float* gp = hw + (size_t)s * HD + c;
  asm volatile("global_load_b128 %0, %1, off th:TH_LOAD_NT_RT\n\t"
               "s_wait_loadcnt 0x0"
               : "=v"(v) : "v"(gp) : "memory");

  float* ap = agg + (size_t)d * HD + c;
  atomicAdd(ap + 0, v.x * w);
  atomicAdd(ap + 1, v.y * w);
  atomicAdd(ap + 2, v.z * w);
  atomicAdd(ap + 3, v.w * w);
}

// out = agg + hw * dinv^2 (self loop) + bias, optional ReLU.
__global__ void finalize_kernel(const float* __restrict__ agg,
                                const float* __restrict__ hw,
                                const float* __restrict__ dinv,
                                const float* __restrict__ bias,
                                float* __restrict__ out, int relu) {
  int t = blockIdx.x * blockDim.x + threadIdx.x;
  if (t >= NT * HD) return;
  const int i = t >> 7;
  const int f = t & 127;
  const float di = dinv[i];
  float v = agg[t] + hw[t] * di * di + bias[f];
  out[t] = relu ? fmaxf(v, 0.0f) : v;
}

// ---------------------------------------------------------------------------
extern "C" void kernel_launch(void* const* d_in, const int* in_sizes, int n_in,
                              void* d_out, int out_size, void* d_ws,
                              size_t ws_size, hipStream_t stream) {
  const float*     x   = (const float*)d_in[0];
  const long long* ei  = (const long long*)d_in[1];   // int64 [2][E]
  // d_in[2] = num_plants (compile-time constant NP)
  const float* pW1 = (const float*)d_in[3];
  const float* pb1 = (const float*)d_in[4];
  const float* pW2 = (const float*)d_in[5];
  const float* pb2 = (const float*)d_in[6];
  const float* qW1 = (const float*)d_in[7];
  const float* qb1 = (const float*)d_in[8];
  const float* qW2 = (const float*)d_in[9];
  const float* qb2 = (const float*)d_in[10];
  const float* gW0 = (const float*)d_in[11];
  const float* gb0 = (const float*)d_in[12];
  const float* gW1 = (const float*)d_in[13];
  const float* gb1 = (const float*)d_in[14];
  float* out = (float*)d_out;

  // workspace layout (~31 MB)
  float* h0   = (float*)d_ws;              // NT*HD
  float* hw   = h0 + (size_t)NT * HD;      // NT*HD
  float* agg  = hw + (size_t)NT * HD;      // NT*HD
  float* deg  = agg + (size_t)NT * HD;     // NT
  float* dinv = deg + NT;                  // NT

  const dim3 B(256);
  const int  gN   = (NT + 255) / 256;
  const int  gE   = (NE + 255) / 256;
  const int  gNH  = (NT * HD + 255) / 256;
  const int  gSc  = (NE * 32 + 255) / 256;

  // symmetric degree normalization (self loop baked in as deg=1)
  fill_kernel<<<gN, B, 0, stream>>>(deg, 1.0f, NT);
  deg_kernel<<<gE, B, 0, stream>>>(ei, deg);
  dinv_kernel<<<gN, B, 0, stream>>>(deg, dinv);

  // node-type MLPs -> h0
  mlp128_kernel<<<NT / 16, B, 0, stream>>>(x, pW1, pb1, pW2, pb2,
                                           qW1, qb1, qW2, qb2, h0);

  // GCN layer 0 (ReLU)
  gemm128_kernel<<<NT / 16, B, 0, stream>>>(h0, gW0, hw);
  fill_kernel<<<gNH, B, 0, stream>>>(agg, 0.0f, NT * HD);
  scatter_kernel<<<gSc, B, 0, stream>>>(ei, hw, dinv, agg);
  finalize_kernel<<<gNH, B, 0, stream>>>(agg, hw, dinv, gb0, h0, 1);

  // GCN layer 1 (no ReLU) -> d_out
  gemm128_kernel<<<NT / 16, B, 0, stream>>>(h0, gW1, hw);
  fill_kernel<<<gNH, B, 0, stream>>>(agg, 0.0f, NT * HD);
  scatter_kernel<<<gSc, B, 0, stream>>>(ei, hw, dinv, agg);
  finalize_kernel<<<gNH, B, 0, stream>>>(agg, hw, dinv, gb1, out, 0);
}